// FlowMPPI_34170759807174
// MI455X (gfx1250) — compile-verified
//
#include <hip/hip_runtime.h>
#include <math.h>

#define HORIZON   100
#define NSAMP     4096
#define NOBS      16
#define NELITE    10
#define DT        0.05f
#define DLOOK_INV 10.0f
#define SIG       0.5f

typedef __attribute__((ext_vector_type(2))) float v2f;
typedef __attribute__((ext_vector_type(8))) float v8f;

__device__ __forceinline__ float clip2(float v) {
    return fminf(fmaxf(v, -2.0f), 2.0f);
}

// ---------------------------------------------------------------------------
// Kernel 1: steering-transform rollout (rollout_si) + total cost, store u_dyn
// one thread per sample; obstacles staged in LDS as v2f (ds_load_b64)
// ---------------------------------------------------------------------------
__global__ void k_rollout_si(const float* __restrict__ state,
                             const float* __restrict__ csin,
                             const float* __restrict__ goal,
                             const float* __restrict__ obst,
                             const float* __restrict__ radp,
                             float* __restrict__ cd,
                             float* __restrict__ costs1) {
    __shared__ v2f lobs[NOBS * HORIZON];           // 12.8 KB
    const v2f* og = (const v2f*)obst;
    for (int i = threadIdx.x; i < NOBS * HORIZON; i += blockDim.x) lobs[i] = og[i];
    __syncthreads();

    int n = blockIdx.x * blockDim.x + threadIdx.x;
    if (n >= NSAMP) return;

    float x = state[0], y = state[1], th = state[2];
    float gx = goal[0], gy = goal[1], rad = radp[0];
    float p0 = 0.f, p1 = 0.f, cost = 0.f;
    const v2f* cs = (const v2f*)csin + (size_t)n * HORIZON;
    v2f* cdr = (v2f*)cd + (size_t)n * HORIZON;

    for (int t = 0; t < HORIZON; ++t) {
        float s, c;
        sincosf(th, &s, &c);
        v2f us = cs[t];
        float v  = us.x * c + us.y * s;
        float wv = (-us.x * s + us.y * c) * DLOOK_INV;
        float u0 = clip2(v), u1 = clip2(wv);

        // stage cost on pre-step state
        float dx = x - gx, dy = y - gy;
        float gc = dx * dx + dy * dy;
        float cc = 0.1f * (u0 * u0 + u1 * u1);
        float e0 = u0 - p0, e1 = u1 - p1;
        float sm = 0.1f * (e0 * e0 + e1 * e1);
        float pen = 0.f;
        for (int o = 0; o < NOBS; ++o) {
            v2f ob = lobs[o * HORIZON + t];
            float ddx = x - ob.x, ddy = y - ob.y;
            float d = sqrtf(ddx * ddx + ddy * ddy);
            float m = fmaxf(rad - d, 0.f);
            pen += m * m;
        }
        cost += gc + cc + sm + 100.f * pen;

        v2f uo; uo.x = u0; uo.y = u1;
        cdr[t] = uo;

        x += u0 * c * DT;
        y += u0 * s * DT;
        th += u1 * DT;
        p0 = u0; p1 = u1;
    }
    float dx = x - gx, dy = y - gy;
    costs1[n] = cost + 10.f * (dx * dx + dy * dy);
}

// ---------------------------------------------------------------------------
// Kernel 2: top-10 (ascending cost, ties -> lower index), destructive on costs1
// ---------------------------------------------------------------------------
__global__ void k_topk(float* __restrict__ costs1, int* __restrict__ elites) {
    __shared__ float sc[256];
    __shared__ int   si[256];
    int tid = threadIdx.x;
    for (int k = 0; k < NELITE; ++k) {
        float best = INFINITY;
        int bidx = 0x7fffffff;
        for (int i = tid; i < NSAMP; i += 256) {
            float cv = costs1[i];
            if (cv < best) { best = cv; bidx = i; }   // ascending scan => lowest idx on tie
        }
        sc[tid] = best; si[tid] = bidx;
        __syncthreads();
        for (int s = 128; s > 0; s >>= 1) {
            if (tid < s) {
                float co = sc[tid + s]; int io = si[tid + s];
                if (co < sc[tid] || (co == sc[tid] && io < si[tid])) {
                    sc[tid] = co; si[tid] = io;
                }
            }
            __syncthreads();
        }
        if (tid == 0) { elites[k] = si[0]; costs1[si[0]] = INFINITY; }
        __syncthreads();
    }
}

// ---------------------------------------------------------------------------
// Kernel 3: perturbed rollout (plain dynamics) + total cost for 10*4096 samples
// pert regenerated on the fly: clip(mean_elite + 0.5*noise). Blocks aligned to
// a single elite (4096/256 = 16 blocks per elite); mean row + obstacles in LDS
// ---------------------------------------------------------------------------
__global__ void k_rollout_pert(const float* __restrict__ state,
                               const float* __restrict__ goal,
                               const float* __restrict__ obst,
                               const float* __restrict__ radp,
                               const float* __restrict__ noise,
                               const float* __restrict__ cd,
                               const int* __restrict__ elites,
                               float* __restrict__ costs2) {
    __shared__ v2f lobs[NOBS * HORIZON];
    __shared__ v2f lmean[HORIZON];
    const v2f* og = (const v2f*)obst;
    for (int i = threadIdx.x; i < NOBS * HORIZON; i += blockDim.x) lobs[i] = og[i];

    int i0 = blockIdx.x * blockDim.x;
    int e  = i0 >> 12;                 // 4096 samples per elite
    int me = elites[e];
    const v2f* mrow = (const v2f*)cd + (size_t)me * HORIZON;
    for (int i = threadIdx.x; i < HORIZON; i += blockDim.x) lmean[i] = mrow[i];
    __syncthreads();

    int idx = i0 + threadIdx.x;
    const v2f* nz = (const v2f*)noise + (size_t)idx * HORIZON;

    float x = state[0], y = state[1], th = state[2];
    float gx = goal[0], gy = goal[1], rad = radp[0];
    float p0 = 0.f, p1 = 0.f, cost = 0.f;

    for (int t = 0; t < HORIZON; ++t) {
        v2f mz = lmean[t];
        v2f nv = nz[t];
        float u0 = clip2(mz.x + SIG * nv.x);
        float u1 = clip2(mz.y + SIG * nv.y);

        float dx = x - gx, dy = y - gy;
        float gc = dx * dx + dy * dy;
        float cc = 0.1f * (u0 * u0 + u1 * u1);
        float e0 = u0 - p0, e1 = u1 - p1;
        float sm = 0.1f * (e0 * e0 + e1 * e1);
        float pen = 0.f;
        for (int o = 0; o < NOBS; ++o) {
            v2f ob = lobs[o * HORIZON + t];
            float ddx = x - ob.x, ddy = y - ob.y;
            float d = sqrtf(ddx * ddx + ddy * ddy);
            float m = fmaxf(rad - d, 0.f);
            pen += m * m;
        }
        cost += gc + cc + sm + 100.f * pen;

        float s, c;
        sincosf(th, &s, &c);
        x += u0 * c * DT;
        y += u0 * s * DT;
        th += u1 * DT;
        p0 = u0; p1 = u1;
    }
    float dx = x - gx, dy = y - gy;
    costs2[idx] = cost + 10.f * (dx * dx + dy * dy);
}

// ---------------------------------------------------------------------------
// Kernel 4: per-elite softmax weights (LAMBDA = 1): min-reduce, exp, sum, norm
// ---------------------------------------------------------------------------
__global__ void k_weights(const float* __restrict__ costs2, float* __restrict__ wbuf) {
    __shared__ float red[256];
    int e = blockIdx.x, tid = threadIdx.x;
    const float* cr = costs2 + (size_t)e * NSAMP;
    float* wr = wbuf + (size_t)e * NSAMP;

    float m = INFINITY;
    for (int n = tid; n < NSAMP; n += 256) m = fminf(m, cr[n]);
    red[tid] = m;
    __syncthreads();
    for (int s = 128; s > 0; s >>= 1) {
        if (tid < s) red[tid] = fminf(red[tid], red[tid + s]);
        __syncthreads();
    }
    float beta = red[0];
    __syncthreads();

    float sum = 0.f;
    for (int n = tid; n < NSAMP; n += 256) {
        float ev = expf(-(cr[n] - beta));
        wr[n] = ev;
        sum += ev;
    }
    red[tid] = sum;
    __syncthreads();
    for (int s = 128; s > 0; s >>= 1) {
        if (tid < s) red[tid] += red[tid + s];
        __syncthreads();
    }
    float inv = 1.f / red[0];
    for (int n = tid; n < NSAMP; n += 256) wr[n] *= inv;
}

// ---------------------------------------------------------------------------
// Kernel 5: elite-weighted blend via V_WMMA_F32_16X16X4_F32.
// Per elite e: opt[e,:] = w[e,:] . pert[e,:,:]  (K = 4096, N = 200 in 16-tiles)
//
// Exploits D = A*B column/row independence: only D row 0 / cols < 200 are
// stored, so A is loaded as an unconditional broadcast of the weight row
// (all 16 A-rows identical) and B's column address is clamped — no exec
// masking anywhere in the loop, EXEC stays all-1s as WMMA requires.
//   A layout: lane holds row lane&15; VGPR0/1 = K {0,1} (lanes 0-15) / {2,3}
//   B layout: lane holds col lane&15; VGPR0/1 = K {0,1} / {2,3}
//   C row M=0 lives in VGPR0 lanes 0-15.
// ---------------------------------------------------------------------------
__global__ void k_wmma_blend(const float* __restrict__ wbuf,
                             const float* __restrict__ cd,
                             const int* __restrict__ elites,
                             const float* __restrict__ noise,
                             float* __restrict__ optbuf) {
    int e    = blockIdx.y;
    int tile = blockIdx.x;
    int lane = threadIdx.x;           // 0..31, wave32
    int row  = lane & 15;
    int kb   = (lane >> 4) << 1;      // 0 (lanes 0-15) or 2 (lanes 16-31)
    int coln = tile * 16 + row;       // output column in 0..199
    bool colok = coln < HORIZON * 2;
    int colc = colok ? coln : (HORIZON * 2 - 1);  // clamped address; OOB cols never stored

    int me = elites[e];
    float mv = cd[(size_t)me * HORIZON * 2 + colc];
    const float* wr = wbuf + (size_t)e * NSAMP;
    const float* nb = noise + (size_t)e * NSAMP * HORIZON * 2 + colc;

    v8f acc = {0.f, 0.f, 0.f, 0.f, 0.f, 0.f, 0.f, 0.f};
    for (int k0 = 0; k0 < NSAMP; k0 += 4) {
        v2f a = *(const v2f*)(wr + k0 + kb);            // b64 broadcast of weight pair
        float n0 = nb[(size_t)(k0 + kb)     * (HORIZON * 2)];
        float n1 = nb[(size_t)(k0 + kb + 1) * (HORIZON * 2)];
        __builtin_prefetch(nb + (size_t)(k0 + 8 + kb) * (HORIZON * 2), 0, 3);
        v2f b;
        b.x = clip2(mv + SIG * n0);
        b.y = clip2(mv + SIG * n1);
        acc = __builtin_amdgcn_wmma_f32_16x16x4_f32(
            /*neg_a=*/false, a, /*neg_b=*/false, b,
            /*c_mod=*/(short)0, acc, /*reuse_a=*/false, /*reuse_b=*/false);
    }
    if (lane < 16 && colok) optbuf[e * HORIZON * 2 + coln] = acc[0];
}

// ---------------------------------------------------------------------------
// Kernel 6: final 10-candidate rollout + cost, argmin, copy winner to output
// ---------------------------------------------------------------------------
__global__ void k_final(const float* __restrict__ state,
                        const float* __restrict__ goal,
                        const float* __restrict__ obst,
                        const float* __restrict__ radp,
                        const float* __restrict__ optbuf,
                        float* __restrict__ out) {
    __shared__ float sc3[NELITE];
    __shared__ int sbest;
    int tid = threadIdx.x;

    if (tid < NELITE) {
        float x = state[0], y = state[1], th = state[2];
        float gx = goal[0], gy = goal[1], rad = radp[0];
        const v2f* u = (const v2f*)optbuf + (size_t)tid * HORIZON;
        const v2f* og = (const v2f*)obst;
        float p0 = 0.f, p1 = 0.f, cost = 0.f;
        for (int t = 0; t < HORIZON; ++t) {
            v2f uv = u[t];
            float u0 = uv.x, u1 = uv.y;
            float dx = x - gx, dy = y - gy;
            float gc = dx * dx + dy * dy;
            float cc = 0.1f * (u0 * u0 + u1 * u1);
            float e0 = u0 - p0, e1 = u1 - p1;
            float sm = 0.1f * (e0 * e0 + e1 * e1);
            float pen = 0.f;
            for (int o = 0; o < NOBS; ++o) {
                v2f ob = og[o * HORIZON + t];
                float ddx = x - ob.x, ddy = y - ob.y;
                float d = sqrtf(ddx * ddx + ddy * ddy);
                float m = fmaxf(rad - d, 0.f);
                pen += m * m;
            }
            cost += gc + cc + sm + 100.f * pen;
            float s, c;
            sincosf(th, &s, &c);
            x += u0 * c * DT;
            y += u0 * s * DT;
            th += u1 * DT;
            p0 = u0; p1 = u1;
        }
        float dx = x - gx, dy = y - gy;
        sc3[tid] = cost + 10.f * (dx * dx + dy * dy);
    }
    __syncthreads();
    if (tid == 0) {
        int b = 0; float bc = sc3[0];
        for (int k = 1; k < NELITE; ++k)
            if (sc3[k] < bc) { bc = sc3[k]; b = k; }
        sbest = b;
    }
    __syncthreads();
    if (tid < HORIZON * 2) out[tid] = optbuf[sbest * HORIZON * 2 + tid];
}

// ---------------------------------------------------------------------------
extern "C" void kernel_launch(void* const* d_in, const int* in_sizes, int n_in,
                              void* d_out, int out_size, void* d_ws, size_t ws_size,
                              hipStream_t stream) {
    (void)in_sizes; (void)n_in; (void)out_size; (void)ws_size;
    const float* state = (const float*)d_in[0];   // (3,)
    const float* csin  = (const float*)d_in[1];   // (4096,100,2)
    /* d_in[2] = horizon (int scalar) -> compile-time HORIZON */
    const float* goal  = (const float*)d_in[3];   // (2,)
    const float* obst  = (const float*)d_in[4];   // (16,100,2)
    const float* radp  = (const float*)d_in[5];   // ()
    const float* noise = (const float*)d_in[6];   // (40960,100,2)
    float* out = (float*)d_out;                   // (100,2)

    float* ws = (float*)d_ws;
    size_t o = 0;
    float* cd     = ws + o; o += (size_t)NSAMP * HORIZON * 2;   // controls_dyn
    float* costs1 = ws + o; o += NSAMP;
    float* wbuf   = ws + o; o += (size_t)NELITE * NSAMP;
    float* costs2 = ws + o; o += (size_t)NELITE * NSAMP;
    float* optbuf = ws + o; o += (size_t)NELITE * HORIZON * 2;
    int* elites   = (int*)(ws + o); o += 16;

    k_rollout_si<<<NSAMP / 256, 256, 0, stream>>>(state, csin, goal, obst, radp, cd, costs1);
    k_topk<<<1, 256, 0, stream>>>(costs1, elites);
    k_rollout_pert<<<(NELITE * NSAMP) / 256, 256, 0, stream>>>(state, goal, obst, radp,
                                                               noise, cd, elites, costs2);
    k_weights<<<NELITE, 256, 0, stream>>>(costs2, wbuf);
    k_wmma_blend<<<dim3((HORIZON * 2 + 15) / 16, NELITE), 32, 0, stream>>>(wbuf, cd, elites,
                                                                           noise, optbuf);
    k_final<<<1, 256, 0, stream>>>(state, goal, obst, radp, optbuf, out);
}